// SoftGBM_88880053223832
// MI455X (gfx1250) — compile-verified
//
#include <hip/hip_runtime.h>
#include <hip/hip_bf16.h>
#include <math.h>

// SoftGBM fused kernel for gfx1250 (MI455X).
// GEMM: logits[b, t*63+n] = dot(x[b,:], W[t,n,:])  via V_WMMA_F32_16X16X4_F32
// then sigmoid -> soft-tree leaf product -> dot(phi) -> *0.1, fully fused.

typedef float v2f __attribute__((ext_vector_type(2)));
typedef float v8f __attribute__((ext_vector_type(8)));
typedef int   v4i __attribute__((vector_size(4 * sizeof(int))));

typedef __attribute__((address_space(1))) v4i* gv4i_ptr;   // global
typedef __attribute__((address_space(3))) v4i* lv4i_ptr;   // LDS

#ifndef __has_builtin
#define __has_builtin(x) 0
#endif

#if __has_builtin(__builtin_amdgcn_global_load_async_to_lds_b128) && \
    __has_builtin(__builtin_amdgcn_s_wait_asynccnt)
#define USE_ASYNC_LDS 1
#else
#define USE_ASYNC_LDS 0
#endif

#define T_NUM_TREES  32
#define T_DEPTH      6
#define T_INPUT_DIM  256
#define T_N_INTERNAL 63
#define T_N_LEAVES   64
#define ROWS_PER_BLOCK 16
#define XSTRIDE 260      // 256 + 4 pad: bank-conflict-free A reads (260 % 64 == 4)
#define PSTRIDE 65       // 64 + 1 pad: bank-conflict-free tree-phase reads
#define WAVES 8
#define TREES_PER_WAVE 4

__global__ __launch_bounds__(256, 2)
void softgbm_wmma_kernel(const float* __restrict__ x,
                         const float* __restrict__ W,
                         const float* __restrict__ bvec,
                         const float* __restrict__ phi,
                         float* __restrict__ out)
{
    __shared__ float xtile[ROWS_PER_BLOCK * XSTRIDE];          // 16.6 KB
    __shared__ float probs[WAVES][ROWS_PER_BLOCK * PSTRIDE];   // 33.3 KB (wave-private slabs)
    __shared__ float outPart[WAVES][ROWS_PER_BLOCK];           // 0.5 KB

    const int tid   = threadIdx.x;
    const int lane  = tid & 31;
    const int wave  = tid >> 5;
    const int lrow  = lane & 15;   // M (A) or N (B/C) within a 16-wide tile
    const int khalf = lane >> 4;   // k sub-offset selector for 16x16x4 fragments
    const int rowBase = blockIdx.x * ROWS_PER_BLOCK;

    // ---- Stage x tile (16 x 256 f32) into LDS ----
    {
        const float* src = x + (size_t)rowBase * T_INPUT_DIM;
        #pragma unroll
        for (int j = 0; j < 4; ++j) {
            int v   = tid + 256 * j;   // float4 index, 1024 total
            int row = v >> 6;          // 64 float4 per row
            int c4  = v & 63;
#if USE_ASYNC_LDS
            __builtin_amdgcn_global_load_async_to_lds_b128(
                (gv4i_ptr)(src + row * T_INPUT_DIM + c4 * 4),
                (lv4i_ptr)(&xtile[row * XSTRIDE + c4 * 4]),
                0, 0);
#else
            const float4 d = *reinterpret_cast<const float4*>(src + row * T_INPUT_DIM + c4 * 4);
            *reinterpret_cast<float4*>(&xtile[row * XSTRIDE + c4 * 4]) = d;
#endif
        }
    }
#if USE_ASYNC_LDS
    __builtin_amdgcn_s_wait_asynccnt(0);
#endif
    __syncthreads();

    float treeAcc = 0.0f;
    float* pw = &probs[wave][0];
    const float* arow = &xtile[lrow * XSTRIDE + 2 * khalf];

    for (int ti = 0; ti < TREES_PER_WAVE; ++ti) {
        const int t = wave * TREES_PER_WAVE + ti;
        const float* Wt = W + (size_t)t * T_N_INTERNAL * T_INPUT_DIM;

        // B-fragment row pointers (node index per lane); pad column 63 clamped to 62.
        const int n3raw = 48 + lrow;
        const float* b0p = Wt + (size_t)(lrow)                          * T_INPUT_DIM + 2 * khalf;
        const float* b1p = Wt + (size_t)(16 + lrow)                     * T_INPUT_DIM + 2 * khalf;
        const float* b2p = Wt + (size_t)(32 + lrow)                     * T_INPUT_DIM + 2 * khalf;
        const float* b3p = Wt + (size_t)((n3raw > 62) ? 62 : n3raw)     * T_INPUT_DIM + 2 * khalf;

        v8f acc0 = {}, acc1 = {}, acc2 = {}, acc3 = {};

        #pragma unroll 4
        for (int k0 = 0; k0 < T_INPUT_DIM; k0 += 4) {
            v2f av = *reinterpret_cast<const v2f*>(arow + k0);   // ds_load_b64 (LDS)
            v2f b0 = *reinterpret_cast<const v2f*>(b0p + k0);    // global_load_b64 (L2-hot)
            v2f b1 = *reinterpret_cast<const v2f*>(b1p + k0);
            v2f b2 = *reinterpret_cast<const v2f*>(b2p + k0);
            v2f b3 = *reinterpret_cast<const v2f*>(b3p + k0);
            acc0 = __builtin_amdgcn_wmma_f32_16x16x4_f32(false, av, false, b0, (short)0, acc0, false, false);
            acc1 = __builtin_amdgcn_wmma_f32_16x16x4_f32(false, av, false, b1, (short)0, acc1, false, false);
            acc2 = __builtin_amdgcn_wmma_f32_16x16x4_f32(false, av, false, b2, (short)0, acc2, false, false);
            acc3 = __builtin_amdgcn_wmma_f32_16x16x4_f32(false, av, false, b3, (short)0, acc3, false, false);
        }

        // ---- bias + sigmoid, scatter C tiles into wave-private LDS probs ----
#define STORE_TILE(ACC, NT) do {                                            \
            int node_ = (NT) * 16 + lrow;                                   \
            if (node_ < T_N_INTERNAL) {                                     \
                float bias_ = bvec[t * T_N_INTERNAL + node_];               \
                _Pragma("unroll")                                           \
                for (int i_ = 0; i_ < 8; ++i_) {                            \
                    int row_ = (lane < 16) ? i_ : (i_ + 8);                 \
                    float z_ = ACC[i_] + bias_;                             \
                    pw[row_ * PSTRIDE + node_] = 1.0f / (1.0f + __expf(-z_)); \
                }                                                           \
            }                                                               \
        } while (0)
        STORE_TILE(acc0, 0);
        STORE_TILE(acc1, 1);
        STORE_TILE(acc2, 2);
        STORE_TILE(acc3, 3);
#undef STORE_TILE

        // LDS ops are in-order within a wave; just stop compiler reordering.
        __builtin_amdgcn_wave_barrier();
        asm volatile("" ::: "memory");

        // ---- soft-tree reduction: lane = (row=lrow, half=khalf) owns 32 leaves ----
        float mu[32];
        {
            float p0 = pw[lrow * PSTRIDE + 0];
            mu[0] = khalf ? p0 : (1.0f - p0);
        }
        #pragma unroll
        for (int d = 1; d <= 5; ++d) {
            const int cnt  = 1 << (d - 1);
            const int base = (1 << d) - 1 + khalf * cnt;
            #pragma unroll
            for (int w = cnt - 1; w >= 0; --w) {
                float p = pw[lrow * PSTRIDE + base + w];
                float m = mu[w];
                mu[2 * w]     = m * (1.0f - p);
                mu[2 * w + 1] = m * p;
            }
        }
        float acc = 0.0f;
        const float* phit = phi + (size_t)t * T_N_LEAVES + khalf * 32;
        #pragma unroll
        for (int w = 0; w < 32; ++w) acc += mu[w] * phit[w];

        acc += __shfl_xor(acc, 16);   // combine the two leaf halves
        treeAcc += acc;

        __builtin_amdgcn_wave_barrier();
        asm volatile("" ::: "memory"); // probs slab reused by next tree
    }

    if (lane < 16) outPart[wave][lrow] = treeAcc;
    __syncthreads();

    if (tid < 16) {
        float s = 0.0f;
        #pragma unroll
        for (int w = 0; w < WAVES; ++w) s += outPart[w][tid];
        out[rowBase + tid] = 0.1f * s;   // SHRINKAGE
    }
}

extern "C" void kernel_launch(void* const* d_in, const int* in_sizes, int n_in,
                              void* d_out, int out_size, void* d_ws, size_t ws_size,
                              hipStream_t stream) {
    const float* x   = (const float*)d_in[0];
    const float* W   = (const float*)d_in[1];
    const float* b   = (const float*)d_in[2];
    const float* phi = (const float*)d_in[3];
    float* out = (float*)d_out;

    const int B = in_sizes[0] / T_INPUT_DIM;          // 16384
    dim3 grid(B / ROWS_PER_BLOCK);                    // 1024 blocks
    dim3 block(256);                                  // 8 wave32
    softgbm_wmma_kernel<<<grid, block, 0, stream>>>(x, W, b, phi, out);
}